// Net_35888746725811
// MI455X (gfx1250) — compile-verified
//
#include <hip/hip_runtime.h>
#include <hip/hip_bf16.h>
#include <math.h>

typedef float v2f __attribute__((ext_vector_type(2)));
typedef float v8f __attribute__((ext_vector_type(8)));

#define N_NODES 100000
#define N_EDGES 3200000
#define IN_FEAT 512
#define HIDDEN  16
#define NCLS    40

#define CHUNK   32            // K-columns staged per chunk
#define NCHUNK  (IN_FEAT / CHUNK)
#define XPAD    36            // 32 + 4 pad floats: row start 16B-aligned, bank-conflict-free

// ---- CDNA5 async copy global->LDS (ASYNCcnt), inline asm: portable across toolchains ----

__device__ __forceinline__ void async_copy16(const float* gsrc, float* lds_dst) {
    unsigned laddr = (unsigned)(unsigned long long)lds_dst;   // low 32 bits = LDS offset
    asm volatile("global_load_async_to_lds_b128 %0, %1, off"
                 :: "v"(laddr), "v"(gsrc) : "memory");
}

__device__ __forceinline__ void wait_async0() {
    asm volatile("s_wait_asynccnt 0x0" ::: "memory");
}

// ---------------- degree / normalization ----------------

__global__ void k_init_deg(float* __restrict__ deg) {
    int i = blockIdx.x * blockDim.x + threadIdx.x;
    if (i < N_NODES) deg[i] = 1.0f;   // self-loop contributes 1
}

__global__ void k_deg_scatter(const long long* __restrict__ row,
                              float* __restrict__ deg) {
    int e = blockIdx.x * blockDim.x + threadIdx.x;
    if (e < N_EDGES) atomicAdd(&deg[(int)row[e]], 1.0f);
}

__global__ void k_deg_to_dinv(float* __restrict__ deg) {
    int i = blockIdx.x * blockDim.x + threadIdx.x;
    if (i < N_NODES) deg[i] = rsqrtf(deg[i]);   // deg >= 1 always
}

// ---------------- layer 1 GEMM: h1 = x @ W1 (WMMA fp32, async-LDS staged) ----------------
// Block = 8 waves = 128 rows. Per 32-col K-chunk, the block async-copies the 128x32
// x tile (padded stride 36) and the 32x16 W1 slice into LDS (double buffered),
// then each wave runs 8 x V_WMMA_F32_16X16X4_F32 on its 16-row tile.
// A frag (16x4): lane L (g=L/16, m=L%16), elem j -> x[m][k+2g+j]
// B frag (4x16): lane L (n=L%16),          elem j -> W1[k+2g+j][n]
// C/D: VGPR r, lane L -> C[r+8g][n]

__device__ __forceinline__ void load_chunk(const float* __restrict__ x,
                                           const float* __restrict__ W1,
                                           int row0, int kk,
                                           float* xs, float* ws, int t) {
    // x tile: 128 rows x 32 cols = 1024 x 16B segments, 4 per thread (coalesced 128B/row)
    #pragma unroll
    for (int i = 0; i < 4; ++i) {
        int s = t + i * 256;
        int r = s >> 3;            // 0..127
        int o = (s & 7) * 4;       // float offset within chunk
        if (row0 + r < N_NODES)
            async_copy16(x + (size_t)(row0 + r) * IN_FEAT + kk + o,
                         xs + r * XPAD + o);
    }
    // W1 chunk: rows kk..kk+31 = 512 floats contiguous = 128 x 16B segments
    if (t < 128)
        async_copy16(W1 + (size_t)kk * HIDDEN + t * 4, ws + t * 4);
}

__global__ void __launch_bounds__(256) k_gemm1_wmma(
        const float* __restrict__ x, const float* __restrict__ W1,
        const float* __restrict__ dinv,
        float* __restrict__ h1, float* __restrict__ agg1) {
    __shared__ float Xs[2][128 * XPAD];       // 2 x 18432 B
    __shared__ float Ws[2][CHUNK * HIDDEN];   // 2 x  2048 B   (Ws[k'][n])

    const int t    = threadIdx.x;
    const int wave = t >> 5;
    const int lane = t & 31;
    const int g    = lane >> 4;
    const int m    = lane & 15;
    const int row0 = blockIdx.x * 128;
    const int row_base = row0 + wave * 16;
    const bool active  = row_base < N_NODES;   // wave-uniform; EXEC stays all-1s inside

    load_chunk(x, W1, row0, 0, Xs[0], Ws[0], t);   // prefetch chunk 0

    v8f acc = {0.f, 0.f, 0.f, 0.f, 0.f, 0.f, 0.f, 0.f};

    for (int c = 0; c < NCHUNK; ++c) {
        wait_async0();        // this wave's chunk-c copies landed in LDS
        __syncthreads();      // all waves' chunk-c visible; prev buffer fully consumed
        if (c + 1 < NCHUNK)   // overlap next chunk's HBM traffic with WMMA
            load_chunk(x, W1, row0, (c + 1) * CHUNK,
                       Xs[(c + 1) & 1], Ws[(c + 1) & 1], t);
        if (active) {
            const float* xa = &Xs[c & 1][(wave * 16 + m) * XPAD + 2 * g];
            const float* wb = &Ws[c & 1][(2 * g) * HIDDEN + m];
            #pragma unroll
            for (int s = 0; s < CHUNK / 4; ++s) {
                v2f a = *(const v2f*)(xa + 4 * s);               // ds_load_b64, conflict-free
                v2f b = { wb[64 * s], wb[64 * s + 16] };          // ds_load_b32 pair
                acc = __builtin_amdgcn_wmma_f32_16x16x4_f32(
                        false, a, false, b, (short)0, acc, false, false);
            }
        }
    }

    if (active) {
        #pragma unroll
        for (int r = 0; r < 8; ++r) {
            int rr  = row_base + r + 8 * g;
            float d = dinv[rr];
            float v = acc[r];
            h1[(size_t)rr * HIDDEN + m]   = v;           // raw transform (gathered later)
            agg1[(size_t)rr * HIDDEN + m] = v * d * d;   // self-loop init of aggregation
        }
    }
}

// ---------------- layer 1 edge scatter ----------------

__global__ void k_scatter1(const long long* __restrict__ row,
                           const long long* __restrict__ col,
                           const float* __restrict__ dinv,
                           const float* __restrict__ h1,
                           float* __restrict__ agg1) {
    long long t = (long long)blockIdx.x * blockDim.x + threadIdx.x;
    if (t >= (long long)N_EDGES * 4) return;
    int e = (int)(t >> 2);
    int q = ((int)t & 3) * 4;
    int r = (int)row[e];
    int c = (int)col[e];
    float nrm = dinv[r] * dinv[c];
    const float4 v = *(const float4*)(h1 + (size_t)c * HIDDEN + q);
    float* dst = agg1 + (size_t)r * HIDDEN + q;
    atomicAdd(dst + 0, v.x * nrm);
    atomicAdd(dst + 1, v.y * nrm);
    atomicAdd(dst + 2, v.z * nrm);
    atomicAdd(dst + 3, v.w * nrm);
}

// ---------------- bias + ReLU in place on agg1 ----------------

__global__ void k_bias_relu(float* __restrict__ agg1,
                            const float* __restrict__ b1) {
    int idx = blockIdx.x * blockDim.x + threadIdx.x;
    if (idx < N_NODES * HIDDEN) {
        float v = agg1[idx] + b1[idx & (HIDDEN - 1)];
        agg1[idx] = v > 0.f ? v : 0.f;
    }
}

// ---------------- layer 2 GEMM (tiny: 16 -> 40) + self-loop init ----------------

__global__ void k_gemm2(const float* __restrict__ h1r,
                        const float* __restrict__ W2,
                        const float* __restrict__ dinv,
                        float* __restrict__ h2,
                        float* __restrict__ out) {
    int idx = blockIdx.x * blockDim.x + threadIdx.x;
    if (idx >= N_NODES * NCLS) return;
    int i = idx / NCLS;
    int c = idx - i * NCLS;
    const float* hv = h1r + (size_t)i * HIDDEN;
    float s = 0.f;
    #pragma unroll
    for (int k = 0; k < HIDDEN; ++k) s += hv[k] * W2[k * NCLS + c];
    h2[idx] = s;
    float d = dinv[i];
    out[idx] = s * d * d;    // self-loop contribution
}

// ---------------- layer 2 edge scatter ----------------

__global__ void k_scatter2(const long long* __restrict__ row,
                           const long long* __restrict__ col,
                           const float* __restrict__ dinv,
                           const float* __restrict__ h2,
                           float* __restrict__ out) {
    long long t = (long long)blockIdx.x * blockDim.x + threadIdx.x;
    if (t >= (long long)N_EDGES * 10) return;
    int e = (int)(t / 10);
    int q = ((int)(t - (long long)e * 10)) * 4;
    int r = (int)row[e];
    int c = (int)col[e];
    float nrm = dinv[r] * dinv[c];
    const float4 v = *(const float4*)(h2 + (size_t)c * NCLS + q);
    float* dst = out + (size_t)r * NCLS + q;
    atomicAdd(dst + 0, v.x * nrm);
    atomicAdd(dst + 1, v.y * nrm);
    atomicAdd(dst + 2, v.z * nrm);
    atomicAdd(dst + 3, v.w * nrm);
}

// ---------------- bias + log_softmax in place on out ----------------

__global__ void k_log_softmax(float* __restrict__ out,
                              const float* __restrict__ b2) {
    int i = blockIdx.x * blockDim.x + threadIdx.x;
    if (i >= N_NODES) return;
    float* o = out + (size_t)i * NCLS;
    float v[NCLS];
    float mx = -INFINITY;
    #pragma unroll
    for (int c = 0; c < NCLS; ++c) {
        v[c] = o[c] + b2[c];
        mx = fmaxf(mx, v[c]);
    }
    float s = 0.f;
    #pragma unroll
    for (int c = 0; c < NCLS; ++c) s += expf(v[c] - mx);
    float lse = mx + logf(s);
    #pragma unroll
    for (int c = 0; c < NCLS; ++c) o[c] = v[c] - lse;
}

// ---------------- launcher ----------------

extern "C" void kernel_launch(void* const* d_in, const int* in_sizes, int n_in,
                              void* d_out, int out_size, void* d_ws, size_t ws_size,
                              hipStream_t stream) {
    const float*     x   = (const float*)d_in[0];
    const long long* ei  = (const long long*)d_in[1];   // int64 [2, E]
    const float*     W1  = (const float*)d_in[2];
    const float*     b1  = (const float*)d_in[3];
    const float*     W2  = (const float*)d_in[4];
    const float*     b2  = (const float*)d_in[5];
    float*           out = (float*)d_out;               // [N, 40]

    const long long* row = ei;
    const long long* col = ei + N_EDGES;

    // workspace carve-up: dinv | h1 | agg1 | h2  (29.2 MB total)
    float* dinv = (float*)d_ws;
    float* h1   = dinv + N_NODES;
    float* agg1 = h1   + (size_t)N_NODES * HIDDEN;
    float* h2   = agg1 + (size_t)N_NODES * HIDDEN;

    const int B = 256;

    // degree -> dinv
    k_init_deg   <<<(N_NODES + B - 1) / B, B, 0, stream>>>(dinv);
    k_deg_scatter<<<(N_EDGES + B - 1) / B, B, 0, stream>>>(row, dinv);
    k_deg_to_dinv<<<(N_NODES + B - 1) / B, B, 0, stream>>>(dinv);

    // layer 1: WMMA transform (async-LDS pipelined) + self-loop init, scatter, bias+relu
    k_gemm1_wmma<<<(N_NODES + 127) / 128, B, 0, stream>>>(x, W1, dinv, h1, agg1);
    long long w1 = (long long)N_EDGES * 4;
    k_scatter1<<<(int)((w1 + B - 1) / B), B, 0, stream>>>(row, col, dinv, h1, agg1);
    k_bias_relu<<<(N_NODES * HIDDEN + B - 1) / B, B, 0, stream>>>(agg1, b1);

    // layer 2: tiny GEMM + self-loop init directly into d_out, scatter, softmax
    k_gemm2<<<(N_NODES * NCLS + B - 1) / B, B, 0, stream>>>(agg1, W2, dinv, h2, out);
    long long w2 = (long long)N_EDGES * 10;
    k_scatter2<<<(int)((w2 + B - 1) / B), B, 0, stream>>>(row, col, dinv, h2, out);
    k_log_softmax<<<(N_NODES + B - 1) / B, B, 0, stream>>>(out, b2);
}